// MeasureDistance_91233695302573
// MI455X (gfx1250) — compile-verified
//
#include <hip/hip_runtime.h>

typedef __attribute__((ext_vector_type(2))) float v2f;
typedef __attribute__((ext_vector_type(8))) float v8f;

#define LOG2E     1.44269504088896340736f
#define TWO_LOG2E 2.88539008177792681472f
#define LN2       0.69314718055994530942f

// c_transform, fused flash-style:
//   out[b,l] = -ln sum_k exp( min(2*x_l.y_k - |x_l|^2 - |y_k|^2, 0) + t[b,k] ),
//   t = pot + logw.
// Stabilized with the wave-computed global shift M = max_k t[b,k] (valid since the
// kernel term is <= 0), so the inner loop needs no online max: one v_exp_f32 plus
// ~4 pk-pairable VALU ops per element. Dot products via V_WMMA_F32_16X16X4_F32
// (D=3 padded to K=4 with zeros). Inner loop is branch-free: the half-wave zero
// padding uses a 0/1 mask multiply (no exec masking) and the prefetch address is
// clamped with min instead of guarded with a branch.
// mode==1: out = 0.5*(result + fold)   (Sinkhorn averaging step)
__global__ __launch_bounds__(128) void ct_kernel(
    const float* __restrict__ xp,   // [B,L,3] row points
    const float* __restrict__ yp,   // [B,K,3] col points
    const float* __restrict__ nx,   // [B,L] row squared norms
    const float* __restrict__ ny,   // [B,K] col squared norms
    const float* __restrict__ pot,  // [B,K] potential over cols
    const float* __restrict__ logw, // [B,K] log weights over cols
    const float* __restrict__ fold, // [B,L] previous potential (mode==1)
    float* __restrict__ out,        // [B,L]
    int L, int K, int totalTiles, int mode)
{
    const int wavesPerBlock = blockDim.x >> 5;
    const int wave = blockIdx.x * wavesPerBlock + (threadIdx.x >> 5);
    if (wave >= totalTiles) return;              // uniform per wave: EXEC stays all-1s

    const int tilesPerBatch = L >> 4;
    const int b       = wave / tilesPerBatch;
    const int rowbase = (wave % tilesPerBatch) << 4;
    const int lane = threadIdx.x & 31;
    const int hh   = lane >> 4;    // half-wave: 0 -> K=0,1 ; 1 -> K=2,pad
    const int li   = lane & 15;
    const float hmask = hh ? 0.f : 1.f;   // zero-pad multiplier for the K=3 slot

    const float* yb   = yp   + (size_t)b * K * 3;
    const float* nyb  = ny   + (size_t)b * K;
    const float* potb = pot  + (size_t)b * K;
    const float* lwb  = logw + (size_t)b * K;

    // ---- wave prepass: M = max_k (pot[k] + logw[k]) over this batch's columns ----
    float Mt = -3.0e38f;
    for (int k = lane; k < K; k += 32)
        Mt = fmaxf(Mt, potb[k] + lwb[k]);
#pragma unroll
    for (int off = 16; off > 0; off >>= 1)
        Mt = fmaxf(Mt, __shfl_xor(Mt, off, 32));
    const float ML2 = Mt * LOG2E;

    // A tile: lane li (half 0) holds x[row].{d0,d1}; half 1 holds {d2, 0}
    const size_t xoff = ((size_t)b * L + rowbase + li) * 3;
    v2f Av;
    Av.x = xp[xoff + (hh ? 2 : 0)];
    Av.y = xp[xoff + 1] * hmask;

    // row squared norms (pre-scaled to base-2) for this lane's 8 C-slots
    float nxrL[8];
#pragma unroll
    for (int j = 0; j < 8; ++j)
        nxrL[j] = nx[(size_t)b * L + rowbase + hh * 8 + j] * LOG2E;

    float s[8];
#pragma unroll
    for (int j = 0; j < 8; ++j) s[j] = 0.f;

    for (int c0 = 0; c0 < K; c0 += 16) {
        const int c = c0 + li;
        const float* yc = yb + (size_t)c * 3;
        v2f Bv;
        Bv.x = yc[hh ? 2 : 0];
        Bv.y = yc[1] * hmask;        // unconditional load + mask: no exec branching

        v8f C = {};
        // D = A(16x4, f32) * B(4x16, f32): exact fp32 pairwise dot products
        C = __builtin_amdgcn_wmma_f32_16x16x4_f32(
                false, Av, false, Bv, (short)0, C, false, false);

        const float u  = nyb[c] * LOG2E;                                // |y_c|^2 log2e
        const float tc = __builtin_fmaf(potb[c] + lwb[c], LOG2E, -ML2); // (t_c - M) log2e

        // branch-free prefetch 4 chunks ahead (clamped) -> global_prefetch_b8
        const int cpf = min(c + 64, K - 1);
        __builtin_prefetch(yb + (size_t)cpf * 3, 0, 1);

#pragma unroll
        for (int j = 0; j < 8; ++j) {
            // (2*dot - |x|^2 - |y|^2) in base-2, clamped at 0, shifted by (t - M)
            float p = __builtin_fmaf(C[j], TWO_LOG2E, -(nxrL[j] + u));
            s[j] += __builtin_amdgcn_exp2f(fminf(p, 0.f) + tc);
        }
    }

    // merge partial sums across the 16 lanes of each half (same 8 rows per half)
#pragma unroll
    for (int off = 1; off < 16; off <<= 1) {
#pragma unroll
        for (int j = 0; j < 8; ++j)
            s[j] += __shfl_xor(s[j], off, 32);
    }

    if (li == 0) {  // lanes 0 and 16 write rows 0-7 / 8-15
#pragma unroll
        for (int j = 0; j < 8; ++j) {
            size_t ridx = (size_t)b * L + rowbase + hh * 8 + j;
            float res = -__builtin_fmaf(__builtin_amdgcn_logf(s[j]), LN2, Mt);
            out[ridx] = mode ? 0.5f * (res + fold[ridx]) : res;
        }
    }
}

__global__ void prep_kernel(const float* __restrict__ x, const float* __restrict__ y,
                            const float* __restrict__ a, const float* __restrict__ bw,
                            float* nx, float* ny, float* la, float* lb,
                            float* f, float* g, float* fx, float* fy,
                            int NL, int NK)
{
    int i = blockIdx.x * blockDim.x + threadIdx.x;
    if (i < NL) {
        float x0 = x[3*i], x1 = x[3*i+1], x2 = x[3*i+2];
        nx[i] = x0*x0 + x1*x1 + x2*x2;
        la[i] = __builtin_amdgcn_logf(a[i]) * LN2;   // natural log
        f[i]  = 0.f; fx[i] = 0.f;
    }
    if (i < NK) {
        float y0 = y[3*i], y1 = y[3*i+1], y2 = y[3*i+2];
        ny[i] = y0*y0 + y1*y1 + y2*y2;
        lb[i] = __builtin_amdgcn_logf(bw[i]) * LN2;
        g[i]  = 0.f; fy[i] = 0.f;
    }
}

// mean_b [ sum_l a*(e1-e3) + sum_k b*(e2-e4) ]
__global__ __launch_bounds__(256) void finalize_kernel(
    const float* __restrict__ a,  const float* __restrict__ bw,
    const float* __restrict__ e1, const float* __restrict__ e2,
    const float* __restrict__ e3, const float* __restrict__ e4,
    float* __restrict__ out, int NL, int NK, float invB)
{
    __shared__ float red[256];
    float p = 0.f;
    for (int i = threadIdx.x; i < NL; i += 256) p += a[i]  * (e1[i] - e3[i]);
    for (int i = threadIdx.x; i < NK; i += 256) p += bw[i] * (e2[i] - e4[i]);
    red[threadIdx.x] = p;
    __syncthreads();
    for (int st = 128; st > 0; st >>= 1) {
        if (threadIdx.x < st) red[threadIdx.x] += red[threadIdx.x + st];
        __syncthreads();
    }
    if (threadIdx.x == 0) out[0] = red[0] * invB;
}

extern "C" void kernel_launch(void* const* d_in, const int* in_sizes, int n_in,
                              void* d_out, int out_size, void* d_ws, size_t ws_size,
                              hipStream_t stream) {
    const float* x  = (const float*)d_in[0];  // [B,L,3]
    const float* a  = (const float*)d_in[1];  // [B,L]
    const float* y  = (const float*)d_in[2];  // [B,K,3]
    const float* bw = (const float*)d_in[3];  // [B,K]

    const int B  = 8;
    const int NL = in_sizes[1];          // B*L
    const int NK = in_sizes[3];          // B*K
    const int L  = NL / B, K = NK / B;

    float* ws = (float*)d_ws;
    float* nx  = ws;        float* ny  = nx  + NL;
    float* la  = ny  + NK;  float* lb  = la  + NL;
    float* f   = lb  + NK;  float* fn  = f   + NL;
    float* g   = fn  + NL;  float* gn  = g   + NK;
    float* fx  = gn  + NK;  float* fxn = fx  + NL;
    float* fy  = fxn + NL;  float* fyn = fy  + NK;
    float* e1  = fyn + NK;  float* e2  = e1  + NL;
    float* e3  = e2  + NK;  float* e4  = e3  + NL;

    int nmax = NL > NK ? NL : NK;
    prep_kernel<<<(nmax + 255) / 256, 256, 0, stream>>>(
        x, y, a, bw, nx, ny, la, lb, f, g, fx, fy, NL, NK);

    const int tilesX = B * (L >> 4);
    const int tilesY = B * (K >> 4);
    dim3 blk(128);
    dim3 grdX((tilesX + 3) / 4), grdY((tilesY + 3) / 4);

    // 20 Jacobi Sinkhorn iterations: fn = ct(g), gn = ct(f), both averaged with olds
    for (int it = 0; it < 20; ++it) {
        ct_kernel<<<grdX, blk, 0, stream>>>(x, y, nx, ny, g,  lb, f,  fn,  L, K, tilesX, 1);
        ct_kernel<<<grdY, blk, 0, stream>>>(y, x, ny, nx, f,  la, g,  gn,  K, L, tilesY, 1);
        float* t;
        t = f;  f  = fn;  fn  = t;
        t = g;  g  = gn;  gn  = t;
        // symmetric (self) potentials for the entropy terms
        ct_kernel<<<grdX, blk, 0, stream>>>(x, x, nx, nx, fx, la, fx, fxn, L, L, tilesX, 1);
        t = fx; fx = fxn; fxn = t;
        ct_kernel<<<grdY, blk, 0, stream>>>(y, y, ny, ny, fy, lb, fy, fyn, K, K, tilesY, 1);
        t = fy; fy = fyn; fyn = t;
    }

    // evaluation transforms (mode=0; fold ptr unused but kept valid)
    ct_kernel<<<grdX, blk, 0, stream>>>(x, y, nx, ny, g,  lb, e1, e1, L, K, tilesX, 0);
    ct_kernel<<<grdY, blk, 0, stream>>>(y, x, ny, nx, f,  la, e2, e2, K, L, tilesY, 0);
    ct_kernel<<<grdX, blk, 0, stream>>>(x, x, nx, nx, fx, la, e3, e3, L, L, tilesX, 0);
    ct_kernel<<<grdY, blk, 0, stream>>>(y, y, ny, ny, fy, lb, e4, e4, K, K, tilesY, 0);

    finalize_kernel<<<1, 256, 0, stream>>>(a, bw, e1, e2, e3, e4,
                                           (float*)d_out, NL, NK, 1.0f / B);
}